// TemporalLoss_89309549953719
// MI455X (gfx1250) — compile-verified
//
#include <hip/hip_runtime.h>
#include <math.h>

// ---- problem constants (from reference) ----
#define NF   3      // frames
#define NS   2      // samples
#define NC   256    // channels
#define HW   128    // spatial
#define PB   256    // 8x8 blocks per image (16*16)
#define NPIX 64     // pixels per block
#define NK   19     // classes
#define LOSS_WEIGHT 1.0f

// LDS row stride (floats). Produced by TDM padding: 64 data dwords + 4 pad
// dwords per channel. bank = (4*row + off) % 64 -> fragment loads conflict-free,
// and rows stay 16-byte aligned.
#define FSTR 68

typedef float    v2f __attribute__((ext_vector_type(2)));
typedef float    v8f __attribute__((ext_vector_type(8)));
typedef unsigned su4 __attribute__((ext_vector_type(4)));
typedef unsigned su8 __attribute__((ext_vector_type(8)));

// One workgroup (256 threads = 8 wave32) per (n, p).
// TDM DMAs one frame's 8x8x256 feature tile into LDS (with hardware padding to
// stride 68); three frames' prototype tiles live in WMMA f32 accumulators;
// 1/||f|| (per pixel) and 1/max(count,1) (per class) are folded into a
// precomputed weighted-onehot LDS matrix so the WMMA loop is pure ds_load+wmma.
__global__ __launch_bounds__(256) void temporal_loss_main(
    const float* __restrict__ feats,
    const int*   __restrict__ masks,
    float*       __restrict__ gacc)   // gacc[0]=sum of loss_sample, gacc[1]=count
{
    __shared__ float fblk[NC * FSTR];    // TDM-staged raw feature tile (one frame)
    __shared__ float bwt[32 * FSTR];     // (mask==k)*invnorm[b]*invcount[k]
    __shared__ float invn[NPIX];         // 1/max(||f||,1e-12) per pixel
    __shared__ int   msk[NPIX];          // mask of current frame's block
    __shared__ float invc[32];           // 1/max(count,1) per class (padded)
    __shared__ int   cnts[NF][32];       // per-frame class counts (padded)
    __shared__ float psum[256];          // norm partial sums
    __shared__ float l1sum[32];          // sum_c |d| per class

    const int tid  = threadIdx.x;
    const int lane = tid & 31;
    const int wave = tid >> 5;
    const int n    = blockIdx.x >> 8;    // sample
    const int p    = blockIdx.x & 255;   // block id: p = bh*16 + bw
    const int by   = (p >> 4) * 8;
    const int bx   = (p & 15) * 8;

    const int half = lane >> 4;          // K-half of the wave
    const int mrow = lane & 15;          // M (A) / N (B,D) index

    v8f acc[NF][2][2];                   // [frame][Mtile-sub][class tile]
    #pragma unroll
    for (int f = 0; f < NF; ++f)
        #pragma unroll
        for (int mt = 0; mt < 2; ++mt)
            #pragma unroll
            for (int ct = 0; ct < 2; ++ct)
                acc[f][mt][ct] = {};

    if (tid < 32) l1sum[tid] = 0.0f;

    // ---- frame-invariant parts of the Tensor DMA descriptor (D#) ----
    // group1: data_size=4B(2), pad_enable, pad_interval=64dw(5), pad_amount=4dw(3),
    //         tile 8 x 8 x 256, tensor_dim0_stride=128, tensor_dim1_stride=16384.
    const unsigned TDBIG = 0x40000000u;  // huge tensor dims: OOB clip never fires
    su8 g1;
    g1[0] = (2u << 16) | (1u << 20) | (5u << 22) | (3u << 25);
    g1[1] = (TDBIG & 0xffffu) << 16;                       // tensor_dim0[15:0]
    g1[2] = (TDBIG >> 16) | ((TDBIG & 0xffffu) << 16);     // td0[31:16] | td1[15:0]
    g1[3] = (TDBIG >> 16) | (8u << 16);                    // td1[31:16] | tile_dim0=8
    g1[4] = 8u | (256u << 16);                             // tile_dim1=8 | tile_dim2=256
    g1[5] = 128u;                                          // stride0[31:0] = 128
    g1[6] = ((16384u & 0xffffu) << 16);                    // stride0[47:32]=0 | stride1[15:0]
    g1[7] = (16384u >> 16);                                // stride1[47:16]
    su4 g2;
    g2[0] = TDBIG;                                         // tensor_dim2
    g2[1] = 0u; g2[2] = 0u; g2[3] = 0u;                    // no dim3, tile_dim3=0
    su4 g3;
    g3[0] = 0u; g3[1] = 0u; g3[2] = 0u; g3[3] = 0u;

    const unsigned lds_fblk = (unsigned)(uintptr_t)(void*)fblk;  // LDS byte offset

    #pragma unroll
    for (int f = 0; f < NF; ++f) {
        const float* fb = feats + ((size_t)(f * NS + n) * NC) * (HW * HW)
                                + (size_t)by * HW + bx;
        const int*   mb = masks + (size_t)(f * NS + n) * (HW * HW)
                                + (size_t)by * HW + bx;

        // -- stage mask (64 ints) --
        if (tid < NPIX) msk[tid] = mb[(tid >> 3) * HW + (tid & 7)];

        // -- TDM: DMA the 8x8x256 tile into fblk with stride-68 padding --
        if (wave == 0) {
            const unsigned long long ga = (unsigned long long)(uintptr_t)fb;
            su4 g0;
            g0[0] = 1u;                                    // count=1 (valid user D#)
            g0[1] = lds_fblk;                              // lds_addr
            g0[2] = (unsigned)(ga & 0xffffffffu);          // global_addr[31:0]
            g0[3] = (unsigned)(ga >> 32) | (2u << 30);     // global_addr[56:32] | type=2
            asm volatile("tensor_load_to_lds %0, %1, %2, %3"
                         :: "s"(g0), "s"(g1), "s"(g2), "s"(g3)
                         : "memory");
            __builtin_amdgcn_s_wait_tensorcnt(0);
        }
        __syncthreads();   // publish DMA'd tile + msk to all waves

        // -- per-pixel sum of squares (4 threads per pixel, 64 channels each) --
        {
            const int b    = tid & 63;
            const int part = tid >> 6;
            float s = 0.0f;
            const int cbase = part * 64;
            for (int i = 0; i < 64; ++i) {
                const float v = fblk[(cbase + i) * FSTR + b];
                s += v * v;
            }
            psum[tid] = s;
        }
        __syncthreads();
        if (tid < NPIX) {
            const float tot = psum[tid] + psum[tid + 64] + psum[tid + 128] + psum[tid + 192];
            invn[tid] = 1.0f / fmaxf(sqrtf(tot), 1e-12f);
        }
        if (tid < NK) {
            int c = 0;
            for (int b2 = 0; b2 < NPIX; ++b2) c += (msk[b2] == tid) ? 1 : 0;
            cnts[f][tid] = c;
            invc[tid] = 1.0f / fmaxf((float)c, 1.0f);
        }
        if (tid >= NK && tid < 32) { invc[tid] = 0.0f; cnts[f][tid] = 0; }
        __syncthreads();

        // -- branchless weighted one-hot: bwt[k][b] = (msk==k)*invn[b]*invc[k] --
        #pragma unroll
        for (int j = 0; j < 8; ++j) {
            const int idx = tid + j * 256;         // 0..2047
            const int k   = idx >> 6;
            const int b   = idx & 63;
            const float sel = (msk[b] == k) ? 1.0f : 0.0f;   // v_cndmask, no branch
            bwt[k * FSTR + b] = sel * invn[b] * invc[k];
        }
        __syncthreads();

        // -- preload B fragments (both class tiles) into registers --
        v2f bfrag[2][16];
        #pragma unroll
        for (int ct = 0; ct < 2; ++ct) {
            const int kbase = (ct * 16 + mrow) * FSTR;
            #pragma unroll
            for (int s = 0; s < 16; ++s) {
                const int p0 = 4 * s + 2 * half;   // K-slot (vgpr j, half h) -> K=2h+j
                bfrag[ct][s].x = bwt[kbase + p0];
                bfrag[ct][s].y = bwt[kbase + p0 + 1];
            }
        }

        // -- WMMA: pure ds_load (A) + 2x wmma per K-step --
        #pragma unroll
        for (int mt = 0; mt < 2; ++mt) {
            const int abase = ((wave * 2 + mt) * 16 + mrow) * FSTR;
            #pragma unroll
            for (int s = 0; s < 16; ++s) {
                const int p0 = 4 * s + 2 * half;
                v2f av;
                av.x = fblk[abase + p0];
                av.y = fblk[abase + p0 + 1];
                #pragma unroll
                for (int ct = 0; ct < 2; ++ct) {
                    acc[f][mt][ct] = __builtin_amdgcn_wmma_f32_16x16x4_f32(
                        false, av, false, bfrag[ct][s], (short)0,
                        acc[f][mt][ct], false, false);
                }
            }
        }
        __syncthreads();   // before next frame reuses LDS
    }

    // -- temporal second difference on the fragments, L1 over channels --
    #pragma unroll
    for (int mt = 0; mt < 2; ++mt) {
        #pragma unroll
        for (int ct = 0; ct < 2; ++ct) {
            const v8f d = acc[0][mt][ct] - 2.0f * acc[1][mt][ct] + acc[2][mt][ct];
            float s = 0.0f;
            #pragma unroll
            for (int r = 0; r < 8; ++r) s += fabsf(d[r]);
            const int kc = ct * 16 + mrow;
            const bool valid = (kc < NK) && (cnts[0][kc] > 0) &&
                               (cnts[1][kc] > 0) && (cnts[2][kc] > 0);
            if (valid) atomicAdd(&l1sum[kc], s);      // ds_add_f32
        }
    }
    __syncthreads();

    if (tid == 0) {
        float pp = 0.0f; int ncls = 0;
        for (int k = 0; k < NK; ++k) {
            if (cnts[0][k] > 0 && cnts[1][k] > 0 && cnts[2][k] > 0) {
                pp += l1sum[k];
                ++ncls;
            }
        }
        if (ncls > 0) {
            // l1 mean over C=256; denom = ncls * (F-2) = ncls * 1
            atomicAdd(&gacc[0], pp / (256.0f * (float)ncls));
            atomicAdd(&gacc[1], 1.0f);
        }
    }
}

__global__ void temporal_loss_zero(float* g) { g[0] = 0.0f; g[1] = 0.0f; }

__global__ void temporal_loss_fin(const float* __restrict__ g, float* __restrict__ out) {
    out[0] = (g[1] > 0.0f) ? (g[0] / g[1]) * LOSS_WEIGHT : 0.0f;
}

extern "C" void kernel_launch(void* const* d_in, const int* in_sizes, int n_in,
                              void* d_out, int out_size, void* d_ws, size_t ws_size,
                              hipStream_t stream) {
    (void)in_sizes; (void)n_in; (void)out_size; (void)ws_size;
    const float* feats = (const float*)d_in[0];
    const int*   masks = (const int*)d_in[1];
    float* gacc = (float*)d_ws;        // 2 floats of scratch, re-zeroed every call

    temporal_loss_zero<<<1, 1, 0, stream>>>(gacc);
    temporal_loss_main<<<NS * PB, 256, 0, stream>>>(feats, masks, gacc);
    temporal_loss_fin<<<1, 1, 0, stream>>>(gacc, (float*)d_out);
}